// BatchOT_33466385171048
// MI455X (gfx1250) — compile-verified
//
#include <hip/hip_runtime.h>
#include <hip/hip_bf16.h>
#include <stdint.h>

// Problem geometry (fixed by the reference: x = (64, 512, 2048) fp32, Q = 256)
#define NDIM   64
#define CDIM   512
#define LDIM   2048
#define MDIM   (NDIM * LDIM)   // 131072 samples per feature channel
#define QDIM   256
#define CHUNK  2048            // one contiguous (n,c) row = 8 KB, sorted in LDS
#define THREADS 256            // 8 wave32 waves per workgroup

typedef __attribute__((ext_vector_type(4))) unsigned int v4u;
typedef __attribute__((ext_vector_type(8))) int          v8i;
typedef __attribute__((ext_vector_type(4))) int          v4i;

// Low 32 bits of a generic pointer into LDS == wave-relative LDS byte offset.
__device__ __forceinline__ unsigned lds_byte_off(const void* p) {
  return (unsigned)(unsigned long long)p;
}

// ---------------------------------------------------------------------------
// CDNA5 Tensor Data Mover: DMA a contiguous run of `nelem` fp32 from global
// into LDS.  D# built per CDNA5 ISA:
//   group0: count=1, lds_addr[63:32], global_addr[120:64], type=2 @127:126
//   group1: data_size=2(4B)@17:16, tensor_dim0@79:48, tensor_dim1=1@111:80,
//           tile_dim0@127:112, tile_dim1=1@143:128, dim0_stride@207:160
// This toolchain exposes the 6-arg builtin:
//   (uint32x4 g0, int32x8 g1, int32x4 g2, int32x4 g3, int32x8 pad, i32 cpol)
// ---------------------------------------------------------------------------
#if __has_builtin(__builtin_amdgcn_tensor_load_to_lds) && __has_builtin(__builtin_amdgcn_s_wait_tensorcnt)
#define HAVE_TDM 1
__device__ __forceinline__ void tdm_load_row_to_lds(const float* gsrc, unsigned ldsoff, int nelem) {
  unsigned long long ga = (unsigned long long)(const void*)gsrc;
  v4u g0;
  g0.x = 1u;                                           // count = 1, user D#
  g0.y = ldsoff;                                       // lds_addr
  g0.z = (unsigned)ga;                                 // global_addr[31:0]
  g0.w = (unsigned)((ga >> 32) & 0x01FFFFFFull) | (2u << 30); // addr[56:32] | type=2
  v8i g1;
  g1[0] = 0x00020000;                                  // data_size = 4 bytes
  g1[1] = (nelem & 0xFFFF) << 16;                      // tensor_dim0[15:0]
  g1[2] = ((nelem >> 16) & 0xFFFF) | (1 << 16);        // tensor_dim0[31:16], tensor_dim1=1
  g1[3] = (nelem & 0xFFFF) << 16;                      // tile_dim0
  g1[4] = 1;                                           // tile_dim1 = 1
  g1[5] = nelem;                                       // tensor_dim0_stride[31:0]
  g1[6] = 0;
  g1[7] = 0;
  v4i gz4 = {0, 0, 0, 0};
  v8i gz8 = {0, 0, 0, 0, 0, 0, 0, 0};
  __builtin_amdgcn_tensor_load_to_lds(g0, g1, gz4, gz4, gz8, 0);
}
#else
#define HAVE_TDM 0
#endif

// CDNA5 async global->LDS load (ASYNCcnt-tracked), GV addressing mode.
__device__ __forceinline__ void async_load_b32(unsigned ldsoff, const void* gaddr) {
  asm volatile("global_load_async_to_lds_b32 %0, %1, off"
               :: "v"(ldsoff), "v"((unsigned long long)gaddr)
               : "memory");
}
__device__ __forceinline__ void wait_asynccnt0() {
  asm volatile("s_wait_asynccnt 0x0" ::: "memory");
}

// ---------------------------------------------------------------------------
// Stage 1: sort each contiguous 2048-float (n,c) row in LDS (bitonic), store
// to the per-column packed layout dst[c*M + n*CHUNK + i].
// ---------------------------------------------------------------------------
__global__ __launch_bounds__(THREADS)
void chunk_sort_kernel(const float* __restrict__ x, float* __restrict__ dst) {
  __shared__ float buf[CHUNK];
  const int bid = blockIdx.x;              // bid = n*CDIM + c
  const int t = threadIdx.x;
  const size_t src = (size_t)bid * CHUNK;

#if HAVE_TDM
  if (t < 32) {                            // wave 0 issues one TDM DMA
    tdm_load_row_to_lds(x + src, lds_byte_off(buf), CHUNK);
    __builtin_amdgcn_s_wait_tensorcnt(0);
  }
#else
  for (int i = t; i < CHUNK; i += THREADS) buf[i] = x[src + i];
#endif
  __syncthreads();

  // Bitonic sort, ascending.  66 passes, disjoint pairs per pass.
  for (int k = 2; k <= CHUNK; k <<= 1) {
    for (int j = k >> 1; j > 0; j >>= 1) {
      for (int i = t; i < CHUNK; i += THREADS) {
        int ixj = i ^ j;
        if (ixj > i) {
          float a = buf[i], b = buf[ixj];
          bool up = ((i & k) == 0);
          if ((a > b) == up) { buf[i] = b; buf[ixj] = a; }
        }
      }
      __syncthreads();
    }
  }

  const int c = bid % CDIM;
  const int n = bid / CDIM;
  float* out = dst + (size_t)c * MDIM + (size_t)n * CHUNK;
  for (int i = t; i < CHUNK; i += THREADS) out[i] = buf[i];
}

// ---------------------------------------------------------------------------
// Stage 2: merge two sorted runs of length S into one of 2S (merge-path).
// One workgroup per pair; each thread co-ranks its output diagonal then
// merges E = 2S/256 elements sequentially (L2-resident 512 KB window).
// ---------------------------------------------------------------------------
__global__ __launch_bounds__(THREADS)
void merge_kernel(const float* __restrict__ src, float* __restrict__ dst, int S) {
  const int p = blockIdx.x;                // pair index within column
  const int c = blockIdx.y;                // feature channel
  const size_t colbase = (size_t)c * MDIM;
  const float* A = src + colbase + (size_t)(2 * p) * S;
  const float* B = A + S;
  float* D = dst + colbase + (size_t)(2 * p) * S;
  const int E = (2 * S) / THREADS;
  const int d0 = threadIdx.x * E;

  // co-rank: i = #elements taken from A among the first d0 outputs (ties: A first)
  int lo = d0 - S; if (lo < 0) lo = 0;
  int hi = d0 < S ? d0 : S;
  while (lo < hi) {
    int mid = (lo + hi) >> 1;
    if (A[mid] <= B[d0 - 1 - mid]) lo = mid + 1; else hi = mid;
  }
  int i = lo, j = d0 - lo;
  __builtin_prefetch(A + i, 0, 0);         // gfx1250 global_prefetch_b8
  __builtin_prefetch(B + j, 0, 0);

  float a = (i < S) ? A[i] : 0.0f;
  float b = (j < S) ? B[j] : 0.0f;
  for (int e = 0; e < E; ++e) {
    bool takeA = (j >= S) || ((i < S) && (a <= b));
    if (takeA) { D[d0 + e] = a; ++i; a = (i < S) ? A[i] : 0.0f; }
    else       { D[d0 + e] = b; ++j; b = (j < S) ? B[j] : 0.0f; }
  }
}

// ---------------------------------------------------------------------------
// Stage 3: extract 256 linear-interpolated order statistics per channel
// (torch/jnp 'linear' quantile: pos = q*(M-1)/(Q-1)).
// ---------------------------------------------------------------------------
__global__ __launch_bounds__(QDIM)
void quantile_kernel(const float* __restrict__ sorted, float* __restrict__ sq) {
  const int c = blockIdx.x, q = threadIdx.x;
  const float* s = sorted + (size_t)c * MDIM;
  double pos = (double)q * (double)(MDIM - 1) / (double)(QDIM - 1);
  int lo = (int)pos;
  if (lo > MDIM - 2) lo = MDIM - 2;
  double g = pos - (double)lo;
  sq[(size_t)c * QDIM + q] = (float)((double)s[lo] + g * ((double)s[lo + 1] - (double)s[lo]));
}

// Sort the 256 target quantiles (one tiny bitonic block).
__global__ __launch_bounds__(QDIM)
void sort_tq_kernel(const float* __restrict__ tq_in, float* __restrict__ tq_out) {
  __shared__ float buf[QDIM];
  const int t = threadIdx.x;
  buf[t] = tq_in[t];
  __syncthreads();
  for (int k = 2; k <= QDIM; k <<= 1) {
    for (int j = k >> 1; j > 0; j >>= 1) {
      int ixj = t ^ j;
      if (ixj > t) {
        float a = buf[t], b = buf[ixj];
        bool up = ((t & k) == 0);
        if ((a > b) == up) { buf[t] = b; buf[ixj] = a; }
      }
      __syncthreads();
    }
  }
  tq_out[t] = buf[t];
}

// ---------------------------------------------------------------------------
// Stage 4: elementwise map.  One block per (n,c) row; async-stage the 256-knot
// source/target tables into LDS, then binary search + two linear interps.
// ---------------------------------------------------------------------------
__global__ __launch_bounds__(THREADS)
void map_kernel(const float* __restrict__ x, const float* __restrict__ sq,
                const float* __restrict__ tq, float* __restrict__ out) {
  __shared__ float s_sq[QDIM];
  __shared__ float s_tq[QDIM];
  const int bid = blockIdx.x;              // n*CDIM + c
  const int c = bid % CDIM;
  const int t = threadIdx.x;

  async_load_b32(lds_byte_off(&s_sq[t]), sq + (size_t)c * QDIM + t);
  async_load_b32(lds_byte_off(&s_tq[t]), tq + t);
  wait_asynccnt0();
  __syncthreads();

  const size_t base = (size_t)bid * LDIM;
  for (int e = 0; e < LDIM / THREADS; ++e) {
    const size_t idx = base + t + e * THREADS;
    float v = x[idx];
    // searchsorted(sq_c, v, 'left'), clipped to [1, Q-1]
    int lo = 0, hi = QDIM;
    while (lo < hi) { int mid = (lo + hi) >> 1; if (s_sq[mid] < v) lo = mid + 1; else hi = mid; }
    int ind = lo < 1 ? 1 : (lo > QDIM - 1 ? QDIM - 1 : lo);
    float x0 = s_sq[ind - 1], x1 = s_sq[ind];
    float u = ((float)(ind - 1) + (v - x0) / (x1 - x0)) * (1.0f / (float)(QDIM - 1));
    u = fminf(fmaxf(u, 0.0f), 1.0f);
    // interp through target quantile fn on uniform knots qs[i] = i/(Q-1)
    float pu = u * (float)(QDIM - 1);
    int i2 = (int)ceilf(pu);
    i2 = i2 < 1 ? 1 : (i2 > QDIM - 1 ? QDIM - 1 : i2);
    float y0 = s_tq[i2 - 1], y1 = s_tq[i2];
    out[idx] = y0 + (y1 - y0) * (pu - (float)(i2 - 1));
  }
}

// ---------------------------------------------------------------------------
// Host: graph-capture-safe launch sequence.  Workspace layout:
//   ws[0 .. M*C)            ping buffer A (256 MB)
//   ws[M*C .. M*C + C*Q)    per-channel source quantiles
//   ws[.. + Q)              sorted target quantiles
// d_out doubles as the pong buffer B (overwritten by the final map pass).
// ---------------------------------------------------------------------------
extern "C" void kernel_launch(void* const* d_in, const int* in_sizes, int n_in,
                              void* d_out, int out_size, void* d_ws, size_t ws_size,
                              hipStream_t stream) {
  const float* x     = (const float*)d_in[0];
  const float* tq_in = (const float*)d_in[1];
  float* out = (float*)d_out;

  float* Abuf  = (float*)d_ws;
  float* sqbuf = Abuf + (size_t)MDIM * CDIM;
  float* tqbuf = sqbuf + (size_t)CDIM * QDIM;
  float* Bbuf  = out;

  chunk_sort_kernel<<<NDIM * CDIM, THREADS, 0, stream>>>(x, Abuf);

  float* src = Abuf;
  float* dst = Bbuf;
  for (int S = CHUNK; S < MDIM; S <<= 1) {        // 6 levels: 2048 -> 131072
    dim3 grid(MDIM / (2 * S), CDIM);
    merge_kernel<<<grid, THREADS, 0, stream>>>(src, dst, S);
    float* tmp = src; src = dst; dst = tmp;
  }
  // 6 swaps => fully sorted columns end up back in Abuf (== src)

  quantile_kernel<<<CDIM, QDIM, 0, stream>>>(src, sqbuf);
  sort_tq_kernel<<<1, QDIM, 0, stream>>>(tq_in, tqbuf);
  map_kernel<<<NDIM * CDIM, THREADS, 0, stream>>>(x, sqbuf, tqbuf, out);
}